// Solver_90958817394764
// MI455X (gfx1250) — compile-verified
//
#include <hip/hip_runtime.h>
#include <hip/hip_bf16.h>
#include <stddef.h>

// ---------------------------------------------------------------------------
// Problem constants (from the reference): L=6 layers, N=4096, Q=M=K2=2048.
// All data fp32. Entirely HBM-bandwidth-bound matvecs; WMMA f32 16x16x4 is
// used for the dominant sequential W@V chain (CDNA5 matrix path, fp32-exact).
// ---------------------------------------------------------------------------
#define NN   4096
#define QQ   2048
#define LL   6

typedef float v2f __attribute__((ext_vector_type(2)));
typedef float v8f __attribute__((ext_vector_type(8)));

// Workspace layout (floats):
//  [0      , 4096 )  Vbuf0
//  [4096   , 8192 )  Vbuf1
//  [8192   , 12288)  WV
//  [12288  , 32768)  PtPi   (5 x 4096)
//  [32768  , 53248)  PhatTpi(5 x 4096)
//  [53248  , 86016)  temp2 partials (8 x 4096)
//  [86016  , 86032)  scalars: [0]=dbl [1]=bsum [2]=gd [3..7]=pi.p [8..12]=s_alpha
#define WS_V0    0
#define WS_V1    4096
#define WS_WV    8192
#define WS_PTPI  12288
#define WS_PHTPI 32768
#define WS_T2P   53248
#define WS_SCAL  86016

// ---------------------------------------------------------------------------
__global__ void k_init(float* __restrict__ scal) {
  if (threadIdx.x < 16) scal[threadIdx.x] = 0.0f;
}

// Phase A: column reductions y[n] = sum_q A[q][n]*s[q]  (all 2048 x 4096).
// jobs 0..4 : P[j]     with pi[j]    -> PtPi[j]
// jobs 5..9 : P_hat[j] with pi[j]    -> PhatTpi[j]
// job  10   : H        with gamma    -> V[l] = -(H^T gamma)
__global__ void __launch_bounds__(256)
k_colred(const float* __restrict__ P, const float* __restrict__ Phat,
         const float* __restrict__ H, const float* __restrict__ pi,
         const float* __restrict__ gamma,
         float* __restrict__ PtPi, float* __restrict__ PhatTpi,
         float* __restrict__ Vl) {
  const int job = blockIdx.y;
  const int n = blockIdx.x * 256 + threadIdx.x;
  const float* A; const float* s; float* out; float sign = 1.0f;
  if (job < 5)       { A = P    + (size_t)job * QQ * NN;     s = pi + job * QQ;     out = PtPi    + job * NN; }
  else if (job < 10) { int j = job - 5;
                       A = Phat + (size_t)j * QQ * NN;       s = pi + j * QQ;       out = PhatTpi + j * NN; }
  else               { A = H;                                s = gamma;             out = Vl; sign = -1.0f; }
  float acc = 0.0f;
  #pragma unroll 4
  for (int q = 0; q < QQ; ++q)
    acc = fmaf(A[(size_t)q * NN + n], s[q], acc);
  out[n] = sign * acc;
}

// Scalar dots: jobs 0..4 -> pidotp[j] = pi[j].p[j]; job 5 -> gd = gamma.d
__global__ void __launch_bounds__(256)
k_dots(const float* __restrict__ pi, const float* __restrict__ p,
       const float* __restrict__ gamma, const float* __restrict__ d,
       float* __restrict__ scal) {
  __shared__ float lds[256];
  const int job = blockIdx.x;
  const float* x; const float* y; float* out;
  if (job < 5) { x = pi + job * QQ; y = p + job * QQ; out = scal + 3 + job; }
  else         { x = gamma;         y = d;            out = scal + 2; }
  float acc = 0.0f;
  for (int q = threadIdx.x; q < QQ; q += 256) acc = fmaf(x[q], y[q], acc);
  lds[threadIdx.x] = acc; __syncthreads();
  for (int s = 128; s > 0; s >>= 1) {
    if (threadIdx.x < s) lds[threadIdx.x] += lds[threadIdx.x + s];
    __syncthreads();
  }
  if (threadIdx.x == 0) *out = lds[0];
}

// bsum += V[l] . b[l-1]   (single block, deterministic tree)
__global__ void __launch_bounds__(256)
k_dot_vb(const float* __restrict__ V, const float* __restrict__ brow,
         float* __restrict__ scal) {
  __shared__ float lds[256];
  float acc = 0.0f;
  for (int n = threadIdx.x; n < NN; n += 256) acc = fmaf(V[n], brow[n], acc);
  lds[threadIdx.x] = acc; __syncthreads();
  for (int s = 128; s > 0; s >>= 1) {
    if (threadIdx.x < s) lds[threadIdx.x] += lds[threadIdx.x + s];
    __syncthreads();
  }
  if (threadIdx.x == 0) scal[1] += lds[0];
}

// ---------------------------------------------------------------------------
// WV = W_i @ v via V_WMMA_F32_16X16X4_F32.
// Block = 4 waves; each block owns 16 output rows, waves split K (4 x 1024).
// A layout (16x4 f32): VGPR0 = K0 (lanes 0-15) / K2 (lanes 16-31), VGPR1 =
// K1 / K3 -> one contiguous v2f load at (row, k + (hi?2:0)).
// B (4x16): vector in column 0. Lanes 0/16 need exactly the SAME (hi?2:0)
// offset pattern, so B = broadcast v2f load masked by a loop-invariant
// bitwise AND (exact zeroing; cannot be re-sunk into a divergent load).
// Software-pipelined: the next 32-k chunk's 16 loads are issued before the
// current chunk's 8 WMMAs consume their registers, so the wave keeps a full
// chunk of loads in flight (partial s_wait_loadcnt instead of drain-to-0).
// D column 0: lane 0 VGPR r = y[m0+r], lane 16 VGPR r = y[m0+8+r].
// ---------------------------------------------------------------------------
__global__ void __launch_bounds__(128)
k_matvec_wmma(const float* __restrict__ Wl, const float* __restrict__ v,
              float* __restrict__ WV) {
  __shared__ float red[4][16];
  const int lane = threadIdx.x & 31;
  const int wave = threadIdx.x >> 5;
  const int m0   = blockIdx.x * 16;
  const int row  = m0 + (lane & 15);
  const int hi   = lane >> 4;                    // 0: lanes 0-15, 1: lanes 16-31
  const unsigned msk = ((lane & 15) == 0) ? 0xFFFFFFFFu : 0u;  // B column 0 only
  const float* wbase = Wl + (size_t)row * NN + (hi ? 2 : 0);
  const float* vbase = v + (hi ? 2 : 0);
  v8f c = {0.f, 0.f, 0.f, 0.f, 0.f, 0.f, 0.f, 0.f};
  const int k0   = wave * 1024;
  const int kend = k0 + 1024;

  v2f a0[8], b0[8];
  v2f a1[8] = {}, b1[8] = {};
  #pragma unroll
  for (int j = 0; j < 8; ++j) {
    a0[j] = *(const v2f*)(wbase + k0 + 4 * j);
    b0[j] = *(const v2f*)(vbase + k0 + 4 * j);   // broadcast within half-wave
  }
  for (int k = k0; k < kend; k += 32) {
    const int kn = k + 32;
    if (kn < kend) {                             // uniform branch
      __builtin_prefetch(wbase + kn + 512, 0, 3);
      #pragma unroll
      for (int j = 0; j < 8; ++j) {
        a1[j] = *(const v2f*)(wbase + kn + 4 * j);
        b1[j] = *(const v2f*)(vbase + kn + 4 * j);
      }
    }
    #pragma unroll
    for (int j = 0; j < 8; ++j) {
      v2f bm;
      bm.x = __uint_as_float(__float_as_uint(b0[j].x) & msk);
      bm.y = __uint_as_float(__float_as_uint(b0[j].y) & msk);
      c = __builtin_amdgcn_wmma_f32_16x16x4_f32(false, a0[j], false, bm,
                                                (short)0, c, false, false);
    }
    #pragma unroll
    for (int j = 0; j < 8; ++j) { a0[j] = a1[j]; b0[j] = b1[j]; }
  }
  if ((lane & 15) == 0) {
    #pragma unroll
    for (int r = 0; r < 8; ++r) red[wave][hi * 8 + r] = c[r];
  }
  __syncthreads();
  if (threadIdx.x < 16) {
    WV[m0 + threadIdx.x] = red[0][threadIdx.x] + red[1][threadIdx.x]
                         + red[2][threadIdx.x] + red[3][threadIdx.x];
  }
}

// s_alpha_i = sum_k alpha[k] * max(-(WV[u_k] - PhatTpi[u_k]), 0)
__global__ void __launch_bounds__(256)
k_salpha(const float* __restrict__ WV, const float* __restrict__ PhatTpi_i,
         const int* __restrict__ uidx, const float* __restrict__ alpha,
         float* __restrict__ outslot) {
  __shared__ float lds[256];
  float acc = 0.0f;
  for (int k = threadIdx.x; k < QQ; k += 256) {
    const int u = uidx[k];
    const float vhat = WV[u] - PhatTpi_i[u];
    acc = fmaf(alpha[k], fmaxf(-vhat, 0.0f), acc);
  }
  lds[threadIdx.x] = acc; __syncthreads();
  for (int s = 128; s > 0; s >>= 1) {
    if (threadIdx.x < s) lds[threadIdx.x] += lds[threadIdx.x + s];
    __syncthreads();
  }
  if (threadIdx.x == 0) *outslot = lds[0];
}

// Per-layer elementwise: builds V[i], accumulates dbl and bsum terms.
__global__ void __launch_bounds__(1024)
k_elem(int layer, const float* __restrict__ WV,
       const float* __restrict__ Lrow, const float* __restrict__ Urow,
       const float* __restrict__ PtPi_i, const float* __restrict__ PhatTpi_i,
       const float* __restrict__ brow, const float* __restrict__ salpha_slot,
       const float* __restrict__ pidotp_slot,
       float* __restrict__ Vout, float* __restrict__ scal) {
  __shared__ float l1[1024];
  __shared__ float l2[1024];
  const float sal = *salpha_slot;
  const float pp  = *pidotp_slot;
  float dbl_acc = 0.0f, b_acc = 0.0f;
  for (int n = threadIdx.x; n < NN; n += 1024) {
    const float Li = Lrow[n], Ui = Urow[n], wv = WV[n];
    const float Ci = (layer == 1 && n == 1) ? 1.0f : 0.0f;
    const float vhat = wv - PhatTpi_i[n];
    const float frac = (fmaxf(vhat, 0.0f) + Ui) / (Ui - Li);
    const bool act   = (Li >= 0.0f);
    const bool deact = (Ui <= 0.0f);
    const bool unst  = (Li < 0.0f) && (Ui > 0.0f);
    const float uV = frac - Ci - sal - PtPi_i[n];
    const float vv = (act ? (wv - Ci) : 0.0f)
                   + (deact ? -Ci : 0.0f)
                   + (unst ? uV : 0.0f);
    Vout[n] = vv;
    if (unst) dbl_acc += frac - pp;
    b_acc = fmaf(vv, brow[n], b_acc);
  }
  l1[threadIdx.x] = dbl_acc; l2[threadIdx.x] = b_acc; __syncthreads();
  for (int s = 512; s > 0; s >>= 1) {
    if (threadIdx.x < s) { l1[threadIdx.x] += l1[threadIdx.x + s];
                           l2[threadIdx.x] += l2[threadIdx.x + s]; }
    __syncthreads();
  }
  if (threadIdx.x == 0) { scal[0] += l1[0]; scal[1] += l2[0]; }
}

// temp2 partials: temp2p[chunk][n] = sum_{m in chunk} W0[m][n] * V1[m]
__global__ void __launch_bounds__(256)
k_temp2(const float* __restrict__ W0, const float* __restrict__ V1,
        float* __restrict__ temp2p) {
  const int n = blockIdx.x * 256 + threadIdx.x;
  const int chunk = blockIdx.y;
  const float* base = W0 + (size_t)chunk * 512 * NN;
  const float* vv = V1 + chunk * 512;
  float acc = 0.0f;
  #pragma unroll 4
  for (int m = 0; m < 512; ++m)
    acc = fmaf(base[(size_t)m * NN + n], vv[m], acc);
  temp2p[(size_t)chunk * NN + n] = acc;
}

// Final: temp2 = -sum_chunks; loss = -(relu(t).L0 - relu(-t).U0 + gd - bsum + dbl)
__global__ void __launch_bounds__(1024)
k_final(const float* __restrict__ temp2p, const float* __restrict__ L0,
        const float* __restrict__ U0, const float* __restrict__ scal,
        float* __restrict__ out) {
  __shared__ float l1[1024];
  __shared__ float l2[1024];
  float s1 = 0.0f, s2 = 0.0f;
  for (int n = threadIdx.x; n < NN; n += 1024) {
    float t = 0.0f;
    #pragma unroll
    for (int c = 0; c < 8; ++c) t += temp2p[c * NN + n];
    t = -t;
    s1 = fmaf(fmaxf(t, 0.0f),  L0[n], s1);
    s2 = fmaf(fmaxf(-t, 0.0f), U0[n], s2);
  }
  l1[threadIdx.x] = s1; l2[threadIdx.x] = s2; __syncthreads();
  for (int s = 512; s > 0; s >>= 1) {
    if (threadIdx.x < s) { l1[threadIdx.x] += l1[threadIdx.x + s];
                           l2[threadIdx.x] += l2[threadIdx.x + s]; }
    __syncthreads();
  }
  if (threadIdx.x == 0)
    out[0] = -(l1[0] - l2[0] + scal[2] - scal[1] + scal[0]);
}

// ---------------------------------------------------------------------------
extern "C" void kernel_launch(void* const* d_in, const int* in_sizes, int n_in,
                              void* d_out, int out_size, void* d_ws, size_t ws_size,
                              hipStream_t stream) {
  (void)in_sizes; (void)n_in; (void)out_size; (void)ws_size;
  const float* gamma = (const float*)d_in[0];
  const float* pi    = (const float*)d_in[1];
  const float* alpha = (const float*)d_in[2];
  const float* W     = (const float*)d_in[3];
  const float* b     = (const float*)d_in[4];
  const float* H     = (const float*)d_in[5];
  const float* dvec  = (const float*)d_in[6];
  const float* Lb    = (const float*)d_in[7];
  const float* Ub    = (const float*)d_in[8];
  const float* P     = (const float*)d_in[9];
  const float* Phat  = (const float*)d_in[10];
  const float* pvec  = (const float*)d_in[11];
  const int*   uidx  = (const int*)d_in[12];
  float* out = (float*)d_out;
  float* ws  = (float*)d_ws;

  float* Vb0    = ws + WS_V0;
  float* Vb1    = ws + WS_V1;
  float* WV     = ws + WS_WV;
  float* PtPi   = ws + WS_PTPI;
  float* PhTpi  = ws + WS_PHTPI;
  float* temp2p = ws + WS_T2P;
  float* scal   = ws + WS_SCAL;

  k_init<<<1, 32, 0, stream>>>(scal);

  dim3 gA(NN / 256, 11);
  k_colred<<<gA, 256, 0, stream>>>(P, Phat, H, pi, gamma, PtPi, PhTpi, Vb0);
  k_dots<<<6, 256, 0, stream>>>(pi, pvec, gamma, dvec, scal);
  k_dot_vb<<<1, 256, 0, stream>>>(Vb0, b + (size_t)(LL - 1) * NN, scal);

  const float* Vin = Vb0;
  float* Vout = Vb1;
  for (int i = LL - 1; i >= 1; --i) {
    k_matvec_wmma<<<NN / 16, 128, 0, stream>>>(W + (size_t)i * NN * NN, Vin, WV);
    k_salpha<<<1, 256, 0, stream>>>(WV, PhTpi + (size_t)(i - 1) * NN,
                                    uidx + (size_t)(i - 1) * QQ,
                                    alpha + (size_t)(i - 1) * QQ,
                                    scal + 8 + (i - 1));
    k_elem<<<1, 1024, 0, stream>>>(i, WV,
                                   Lb + (size_t)i * NN, Ub + (size_t)i * NN,
                                   PtPi  + (size_t)(i - 1) * NN,
                                   PhTpi + (size_t)(i - 1) * NN,
                                   b + (size_t)(i - 1) * NN,
                                   scal + 8 + (i - 1), scal + 3 + (i - 1),
                                   Vout, scal);
    const float* t = Vin; Vin = Vout; Vout = (float*)t;
  }
  // Vin == V[1]
  dim3 gT(NN / 256, 8);
  k_temp2<<<gT, 256, 0, stream>>>(W, Vin, temp2p);
  k_final<<<1, 1024, 0, stream>>>(temp2p, Lb, Ub, scal, out);
}